// HeteroDoshaNet_8976481648800
// MI455X (gfx1250) — compile-verified
//
#include <hip/hip_runtime.h>
#include <math.h>

typedef __attribute__((ext_vector_type(2))) float v2f;
typedef __attribute__((ext_vector_type(8))) float v8f;

#define NPAT 100000
#define EST  300000
#define HID  128
#define HEADS 4

// ---------- helpers ----------
__global__ void fill_f(float* __restrict__ p, float v, int n) {
  int i = blockIdx.x * blockDim.x + threadIdx.x;
  if (i < n) p[i] = v;
}

__device__ __forceinline__ void atomicMaxF(float* addr, float v) {
  // monotone bit trick: int-max for >=0, uint-min for <0 (init must be -inf)
  if (v >= 0.0f) atomicMax((int*)addr, __float_as_int(v));
  else           atomicMin((unsigned int*)addr, __float_as_uint(v));
}

// ---------- layer 1 GEMM: h1 = x[100000,64] @ W1[64,128] + b1 ----------
// grid.x = 6250 M-tiles (16 rows each); block = 256 = 8 waves; wave w owns
// N-strip [16w, 16w+16). Uses V_WMMA_F32_16X16X4_F32, 16 k-steps of K=4.
// A frag (16x4 f32, 2 VGPR): lanes 0-15 hold M=lane K={4k,4k+1}; lanes 16-31
// hold M=lane-16 K={4k+2,4k+3}. B frag mirrors: v0/v1 = rows {4k+2h, 4k+2h+1}.
__global__ void gemm1_wmma(const float* __restrict__ X,
                           const float* __restrict__ W,
                           const float* __restrict__ bias,
                           float* __restrict__ H) {
  int wave = threadIdx.x >> 5;
  int lane = threadIdx.x & 31;
  int lhalf = lane >> 4;         // 0 or 1
  int l = lane & 15;
  int row0 = blockIdx.x * 16;
  int col = wave * 16 + l;       // output / B column
  v8f acc = {};
  const float* arow = X + (size_t)(row0 + l) * 64 + 2 * lhalf;
#pragma unroll
  for (int k = 0; k < 16; ++k) {
    int kb = 4 * k;
    v2f a; a.x = arow[kb]; a.y = arow[kb + 1];
    int kr = kb + 2 * lhalf;
    v2f b; b.x = W[kr * HID + col]; b.y = W[(kr + 1) * HID + col];
    acc = __builtin_amdgcn_wmma_f32_16x16x4_f32(
        false, a, false, b, (short)0, acc, false, false);
  }
  float bv = bias[col];
#pragma unroll
  for (int i = 0; i < 8; ++i) {
    int r = row0 + i + 8 * lhalf;           // C/D layout: vgpr i -> M=i / i+8
    H[(size_t)r * HID + col] = acc[i] + bv;
  }
}

// ---------- per-node attention dots (layer 1, 4 heads x 32 dims) ----------
__global__ void dots1(const float* __restrict__ H,
                      const float* __restrict__ ats,   // (4,32)
                      const float* __restrict__ atd,   // (4,32)
                      float* __restrict__ AS, float* __restrict__ AD) {
  int idx = blockIdx.x * blockDim.x + threadIdx.x;
  if (idx >= NPAT * HEADS) return;
  int node = idx >> 2, h = idx & 3;
  const float* p  = H + (size_t)node * HID + h * 32;
  const float* ws = ats + h * 32;
  const float* wd = atd + h * 32;
  float s = 0.f, d = 0.f;
#pragma unroll
  for (int j = 0; j < 32; ++j) { float v = p[j]; s += v * ws[j]; d += v * wd[j]; }
  AS[idx] = s; AD[idx] = d;
}

// ---------- edge softmax pass 1: leaky-relu logits + segment max ----------
__global__ void edge_max1(const int* __restrict__ src, const int* __restrict__ dst,
                          const float* __restrict__ AS, const float* __restrict__ AD,
                          float* __restrict__ E, float* __restrict__ M) {
  int idx = blockIdx.x * blockDim.x + threadIdx.x;
  if (idx >= EST * HEADS) return;
  int e = idx >> 2, h = idx & 3;
  int s = src[e], d = dst[e];
  float al = AS[s * 4 + h] + AD[d * 4 + h];
  al = al > 0.f ? al : 0.2f * al;
  E[idx] = al;
  atomicMaxF(&M[d * 4 + h], al);
}

// ---------- edge softmax pass 2: exp + segment sum ----------
__global__ void edge_exp1(const int* __restrict__ dst,
                          float* __restrict__ E, const float* __restrict__ M,
                          float* __restrict__ S) {
  int idx = blockIdx.x * blockDim.x + threadIdx.x;
  if (idx >= EST * HEADS) return;
  int e = idx >> 2, h = idx & 3;
  int d = dst[e];
  float ex = expf(E[idx] - M[d * 4 + h]);
  E[idx] = ex;
  atomicAdd(&S[d * 4 + h], ex);
}

// ---------- message aggregation: agg[dst,c] += h1[src,c]*alpha[e,head(c)] --
__global__ void edge_agg1(const int* __restrict__ src, const int* __restrict__ dst,
                          const float* __restrict__ E, const float* __restrict__ S,
                          const float* __restrict__ H, float* __restrict__ AGG) {
  int idx = blockIdx.x * blockDim.x + threadIdx.x;
  if (idx >= EST * HID) return;
  int e = idx >> 7, c = idx & 127, h = c >> 5;
  int s = src[e], d = dst[e];
  float alpha = E[e * 4 + h] / S[d * 4 + h];
  atomicAdd(&AGG[(size_t)d * HID + c], H[(size_t)s * HID + c] * alpha);
}

// ---------- BN stats over relu(agg): per-channel sum & sumsq ----------
__global__ void bn_stats(const float* __restrict__ AGG,
                         float* __restrict__ SUM, float* __restrict__ SQ) {
  int c = threadIdx.x;               // 128 threads
  int r0 = blockIdx.x * 250;         // 400 blocks * 250 rows
  float s = 0.f, q = 0.f;
  for (int r = r0; r < r0 + 250; ++r) {
    float v = AGG[(size_t)r * HID + c];
    v = v > 0.f ? v : 0.f;
    s += v; q += v * v;
  }
  atomicAdd(&SUM[c], s);
  atomicAdd(&SQ[c], q);
}

// ---------- relu -> BN -> ELU, writes back into H (h1 no longer needed) ----
__global__ void bn_apply(const float* __restrict__ AGG,
                         const float* __restrict__ SUM, const float* __restrict__ SQ,
                         const float* __restrict__ gamma, const float* __restrict__ beta,
                         float* __restrict__ H) {
  int idx = blockIdx.x * blockDim.x + threadIdx.x;
  if (idx >= NPAT * HID) return;
  int c = idx & 127;
  float mean = SUM[c] * (1.0f / NPAT);
  float var  = SQ[c] * (1.0f / NPAT) - mean * mean;   // biased var (matches ref)
  float r = AGG[idx];
  r = r > 0.f ? r : 0.f;
  float xh = (r - mean) * rsqrtf(var + 1e-5f);
  float y = gamma[c] * xh + beta[c];
  H[idx] = y > 0.f ? y : expm1f(y);                   // ELU(alpha=1)
}

// ---------- layer 2: h2 = hbn @ W2[128,3] + b2, plus attention dots -------
// one wave per node; lane owns 4 channels (float4, coalesced 512B row read)
__global__ void gemm2_dots(const float* __restrict__ HB,
                           const float* __restrict__ W2, const float* __restrict__ b2,
                           const float* __restrict__ at2s, const float* __restrict__ at2d,
                           float* __restrict__ H2, float* __restrict__ A2S,
                           float* __restrict__ A2D) {
  __shared__ float w[HID * 3];
  for (int i = threadIdx.x; i < HID * 3; i += blockDim.x) w[i] = W2[i];
  __syncthreads();
  int wave = threadIdx.x >> 5, lane = threadIdx.x & 31;
  int node = blockIdx.x * 8 + wave;     // grid 12500, 8 waves/block
  const float4 v = *(const float4*)(HB + (size_t)node * HID + lane * 4);
  int cb = lane * 4;
  float p0 = v.x * w[(cb + 0) * 3 + 0] + v.y * w[(cb + 1) * 3 + 0] +
             v.z * w[(cb + 2) * 3 + 0] + v.w * w[(cb + 3) * 3 + 0];
  float p1 = v.x * w[(cb + 0) * 3 + 1] + v.y * w[(cb + 1) * 3 + 1] +
             v.z * w[(cb + 2) * 3 + 1] + v.w * w[(cb + 3) * 3 + 1];
  float p2 = v.x * w[(cb + 0) * 3 + 2] + v.y * w[(cb + 1) * 3 + 2] +
             v.z * w[(cb + 2) * 3 + 2] + v.w * w[(cb + 3) * 3 + 2];
#pragma unroll
  for (int off = 16; off > 0; off >>= 1) {
    p0 += __shfl_xor(p0, off, 32);
    p1 += __shfl_xor(p1, off, 32);
    p2 += __shfl_xor(p2, off, 32);
  }
  if (lane == 0) {
    float h0 = p0 + b2[0], h1 = p1 + b2[1], h2 = p2 + b2[2];
    H2[node * 3 + 0] = h0; H2[node * 3 + 1] = h1; H2[node * 3 + 2] = h2;
    A2S[node] = h0 * at2s[0] + h1 * at2s[1] + h2 * at2s[2];
    A2D[node] = h0 * at2d[0] + h1 * at2d[1] + h2 * at2d[2];
  }
}

// ---------- layer 2 edge softmax (1 head) ----------
__global__ void edge_max2(const int* __restrict__ src, const int* __restrict__ dst,
                          const float* __restrict__ AS, const float* __restrict__ AD,
                          float* __restrict__ E, float* __restrict__ M) {
  int e = blockIdx.x * blockDim.x + threadIdx.x;
  if (e >= EST) return;
  float al = AS[src[e]] + AD[dst[e]];
  al = al > 0.f ? al : 0.2f * al;
  E[e] = al;
  atomicMaxF(&M[dst[e]], al);
}

__global__ void edge_exp2(const int* __restrict__ dst, float* __restrict__ E,
                          const float* __restrict__ M, float* __restrict__ S) {
  int e = blockIdx.x * blockDim.x + threadIdx.x;
  if (e >= EST) return;
  int d = dst[e];
  float ex = expf(E[e] - M[d]);
  E[e] = ex;
  atomicAdd(&S[d], ex);
}

__global__ void edge_agg2(const int* __restrict__ src, const int* __restrict__ dst,
                          const float* __restrict__ E, const float* __restrict__ S,
                          const float* __restrict__ H2, float* __restrict__ O2) {
  int idx = blockIdx.x * blockDim.x + threadIdx.x;
  if (idx >= EST * 3) return;
  int e = idx / 3, c = idx % 3;
  int s = src[e], d = dst[e];
  float alpha = E[e] / S[d];
  atomicAdd(&O2[d * 3 + c], H2[s * 3 + c] * alpha);
}

// ---------- relu + log_softmax over 3 classes ----------
__global__ void final_lsm(const float* __restrict__ O2, float* __restrict__ out) {
  int n = blockIdx.x * blockDim.x + threadIdx.x;
  if (n >= NPAT) return;
  float x0 = fmaxf(O2[n * 3 + 0], 0.f);
  float x1 = fmaxf(O2[n * 3 + 1], 0.f);
  float x2 = fmaxf(O2[n * 3 + 2], 0.f);
  float m = fmaxf(x0, fmaxf(x1, x2));
  float lse = m + logf(expf(x0 - m) + expf(x1 - m) + expf(x2 - m));
  out[n * 3 + 0] = x0 - lse;
  out[n * 3 + 1] = x1 - lse;
  out[n * 3 + 2] = x2 - lse;
}

extern "C" void kernel_launch(void* const* d_in, const int* in_sizes, int n_in,
                              void* d_out, int out_size, void* d_ws, size_t ws_size,
                              hipStream_t stream) {
  // Recursive insertion-order flattening of setup_inputs():
  const float* x_p    = (const float*)d_in[0];   // (100000,64)
  const int*   st_src = (const int*)d_in[7];     // (300000,)
  const int*   st_dst = (const int*)d_in[8];     // (300000,)
  const float* W1     = (const float*)d_in[9];   // han1.lin.patient.W (64,128)
  const float* b1     = (const float*)d_in[10];  // han1.lin.patient.b (128,)
  const float* at1s   = (const float*)d_in[19];  // han1.att.similar_to.src (4,32)
  const float* at1d   = (const float*)d_in[20];  // han1.att.similar_to.dst (4,32)
  const float* gamma  = (const float*)d_in[24];  // bn.gamma (128,)
  const float* beta   = (const float*)d_in[25];  // bn.beta  (128,)
  const float* W2     = (const float*)d_in[26];  // han2.lin.patient.W (128,3)
  const float* b2     = (const float*)d_in[27];  // han2.lin.patient.b (3,)
  const float* at2s   = (const float*)d_in[36];  // han2.att.similar_to.src (1,3)
  const float* at2d   = (const float*)d_in[37];  // han2.att.similar_to.dst (1,3)
  float* out = (float*)d_out;

  // workspace layout (floats)
  float* base = (float*)d_ws;
  float* H1   = base;                                // 12.8M (later reused as hbn)
  float* AGG1 = H1   + (size_t)NPAT * HID;           // 12.8M
  float* AS1  = AGG1 + (size_t)NPAT * HID;           // 400k
  float* AD1  = AS1  + NPAT * HEADS;                 // 400k
  float* M1   = AD1  + NPAT * HEADS;                 // 400k
  float* S1   = M1   + NPAT * HEADS;                 // 400k
  float* E1   = S1   + NPAT * HEADS;                 // 1.2M
  float* BNS  = E1   + (size_t)EST * HEADS;          // 128
  float* BNQ  = BNS  + HID;                          // 128
  float* H2   = BNQ  + HID;                          // 300k
  float* A2S  = H2   + NPAT * 3;                     // 100k
  float* A2D  = A2S  + NPAT;                         // 100k
  float* M2   = A2D  + NPAT;                         // 100k
  float* S2   = M2   + NPAT;                         // 100k
  float* E2   = S2   + NPAT;                         // 300k
  float* O2   = E2   + EST;                          // 300k

  const int T = 256;
  auto nb = [](int n, int t) { return (n + t - 1) / t; };

  // per-call re-init (graph-replay deterministic)
  fill_f<<<nb(NPAT * HID, T), T, 0, stream>>>(AGG1, 0.f, NPAT * HID);
  fill_f<<<nb(NPAT * HEADS, T), T, 0, stream>>>(S1, 0.f, NPAT * HEADS);
  fill_f<<<nb(NPAT * HEADS, T), T, 0, stream>>>(M1, -INFINITY, NPAT * HEADS);
  fill_f<<<1, 128, 0, stream>>>(BNS, 0.f, HID);
  fill_f<<<1, 128, 0, stream>>>(BNQ, 0.f, HID);
  fill_f<<<nb(NPAT, T), T, 0, stream>>>(M2, -INFINITY, NPAT);
  fill_f<<<nb(NPAT, T), T, 0, stream>>>(S2, 0.f, NPAT);
  fill_f<<<nb(NPAT * 3, T), T, 0, stream>>>(O2, 0.f, NPAT * 3);

  // layer 1
  gemm1_wmma<<<NPAT / 16, 256, 0, stream>>>(x_p, W1, b1, H1);
  dots1<<<nb(NPAT * HEADS, T), T, 0, stream>>>(H1, at1s, at1d, AS1, AD1);
  edge_max1<<<nb(EST * HEADS, T), T, 0, stream>>>(st_src, st_dst, AS1, AD1, E1, M1);
  edge_exp1<<<nb(EST * HEADS, T), T, 0, stream>>>(st_dst, E1, M1, S1);
  edge_agg1<<<nb(EST * HID, T), T, 0, stream>>>(st_src, st_dst, E1, S1, H1, AGG1);
  bn_stats<<<NPAT / 250, HID, 0, stream>>>(AGG1, BNS, BNQ);
  bn_apply<<<nb(NPAT * HID, T), T, 0, stream>>>(AGG1, BNS, BNQ, gamma, beta, H1);

  // layer 2
  gemm2_dots<<<NPAT / 8, 256, 0, stream>>>(H1, W2, b2, at2s, at2d, H2, A2S, A2D);
  edge_max2<<<nb(EST, T), T, 0, stream>>>(st_src, st_dst, A2S, A2D, E2, M2);
  edge_exp2<<<nb(EST, T), T, 0, stream>>>(st_dst, E2, M2, S2);
  edge_agg2<<<nb(EST * 3, T), T, 0, stream>>>(st_src, st_dst, E2, S2, H2, O2);
  final_lsm<<<nb(NPAT, T), T, 0, stream>>>(O2, out);
}